// LengthRegulator_38431367364689
// MI455X (gfx1250) — compile-verified
//
#include <hip/hip_runtime.h>

// FastSpeech LengthRegulator for MI455X (gfx1250).
// Memory-bound gather/pad: no matmul => WMMA is inapplicable. The roofline
// is HBM bandwidth: 192MB output write + ~24MB gather read (x fits in the
// 192MB L2 and is re-read ~3.5x, so gather re-reads hit L2) => ~9.3us floor
// at 23.3 TB/s. Design: b128 vector memory ops, coalesced addressing, and
// CDNA5 non-temporal store hints (TH=NT) on the streamed output so it does
// not evict the hot x working set from L2.
//
// Shapes fixed by the reference setup: B=32, T=512, C=384.
// mel_max_len (M) is recovered on the host from out_size = B*M*C + B.

typedef float float4v __attribute__((ext_vector_type(4)));

static constexpr int B_  = 32;
static constexpr int T_  = 512;
static constexpr int C_  = 384;
static constexpr int C4_ = C_ / 4;    // 96 float4 per row
static constexpr int NWAVE_ = T_ / 32; // 16 wave32 per scan block

// ---------------------------------------------------------------------------
// Kernel 1: per-batch inclusive scan of duration[b, 0..T).
// Wave32 shuffle scan (lane-permute, no LDS traffic, no barriers) + one
// 16-element cross-wave combine => only 2 workgroup barriers total.
// Writes csum (int, for kernels 2/3), and mel_len as float into the tail of
// d_out (tuple output #2 in d_out's dtype).
// ---------------------------------------------------------------------------
__global__ void lr_scan(const int* __restrict__ dur,
                        int* __restrict__ csum,
                        int* __restrict__ mel_len_i,
                        float* __restrict__ mel_len_f) {
  __shared__ int wave_sum[NWAVE_];
  const int b    = blockIdx.x;
  const int t    = threadIdx.x;     // 0..511
  const int lane = t & 31;
  const int wave = t >> 5;

  int v = dur[b * T_ + t];

  // Intra-wave inclusive scan: 5 shuffle steps on wave32.
#pragma unroll
  for (int off = 1; off < 32; off <<= 1) {
    int n = __shfl_up(v, off, 32);
    if (lane >= off) v += n;
  }

  if (lane == 31) wave_sum[wave] = v;
  __syncthreads();

  // Wave 0 scans the 16 wave totals (again pure shuffles).
  if (wave == 0) {
    int s = (lane < NWAVE_) ? wave_sum[lane] : 0;
#pragma unroll
    for (int off = 1; off < NWAVE_; off <<= 1) {
      int n = __shfl_up(s, off, 32);
      if (lane >= off) s += n;
    }
    if (lane < NWAVE_) wave_sum[lane] = s;
  }
  __syncthreads();

  if (wave > 0) v += wave_sum[wave - 1];

  csum[b * T_ + t] = v;
  if (t == T_ - 1) {
    mel_len_i[b] = v;
    mel_len_f[b] = (float)v;   // tuple output #2
  }
}

// ---------------------------------------------------------------------------
// Kernel 2: tok[b, m] = searchsorted(csum[b], m, side='right'), clamped to
// T-1; -1 sentinel for padded frames (m >= mel_len). Branchless uniform
// binary descent: 9 v_cndmask steps over the 2KB per-batch csum (cache-hot).
// The descent caps at T-1 = 511 by construction, matching the reference
// clamp; count==512 only occurs for m >= mel_len, already handled.
// ---------------------------------------------------------------------------
__global__ void lr_tok(const int* __restrict__ csum,
                       const int* __restrict__ mel_len_i,
                       int* __restrict__ tok,
                       int M) {
  const int b = blockIdx.y;
  const int m = blockIdx.x * blockDim.x + threadIdx.x;
  if (m >= M) return;
  if (m >= mel_len_i[b]) { tok[b * M + m] = -1; return; }
  const int* __restrict__ c = csum + b * T_;
  int cnt = 0;
#pragma unroll
  for (int step = T_ / 2; step > 0; step >>= 1) {
    cnt += (c[cnt + step - 1] <= m) ? step : 0;
  }
  tok[b * M + m] = cnt;   // cnt <= T-1 here by construction
}

// ---------------------------------------------------------------------------
// Kernel 3: one thread per output float4. 96 consecutive lanes share one
// tok[row] load (merged broadcast), gather x[b, tok] with coalesced b128
// loads (L2-hot), and stream the result out with a non-temporal b128 store.
// ---------------------------------------------------------------------------
__global__ void lr_gather(const float4v* __restrict__ x,   // [B, T, C4]
                          const int* __restrict__ tok,     // [B, M]
                          float4v* __restrict__ out,       // [B, M, C4]
                          int M) {
  const int b   = blockIdx.y;
  const int idx = blockIdx.x * blockDim.x + threadIdx.x;   // in [0, M*C4)
  if (idx >= M * C4_) return;
  const int row = idx / C4_;            // magic-number div by 96
  const int c4  = idx - row * C4_;
  const int tk  = tok[b * M + row];
  float4v v = {0.f, 0.f, 0.f, 0.f};
  if (tk >= 0) {
    v = x[((size_t)b * T_ + tk) * C4_ + c4];               // global_load_b128
  }
  __builtin_nontemporal_store(v, &out[(size_t)b * M * C4_ + idx]);  // TH=NT
}

// ---------------------------------------------------------------------------
extern "C" void kernel_launch(void* const* d_in, const int* in_sizes, int n_in,
                              void* d_out, int out_size, void* d_ws, size_t ws_size,
                              hipStream_t stream) {
  const float* x   = (const float*)d_in[0];       // [B, T, C] f32
  const int*   dur = (const int*)d_in[1];         // [B, T] (JAX x64 off => i32)
  (void)in_sizes; (void)n_in; (void)ws_size;

  // Recover mel_max_len from the flattened tuple output size: B*M*C + B.
  const int M = (out_size - B_) / (B_ * C_);

  // Workspace layout (ints): csum[B*T] | mel_len[B] | tok[B*M]  (~580 KB)
  int* ws        = (int*)d_ws;
  int* csum      = ws;
  int* mel_len_i = csum + B_ * T_;
  int* tok       = mel_len_i + B_;

  float* mel_out   = (float*)d_out;
  float* mel_len_f = mel_out + (size_t)B_ * M * C_;

  lr_scan<<<B_, T_, 0, stream>>>(dur, csum, mel_len_i, mel_len_f);

  dim3 g2((M + 255) / 256, B_);
  lr_tok<<<g2, 256, 0, stream>>>(csum, mel_len_i, tok, M);

  dim3 g3((M * C4_ + 255) / 256, B_);
  lr_gather<<<g3, 256, 0, stream>>>((const float4v*)x, tok,
                                    (float4v*)d_out, M);
}